// RelationAwareGNN_56599079026738
// MI455X (gfx1250) — compile-verified
//
#include <hip/hip_runtime.h>

constexpr int D = 256;          // input dim
constexpr int H = 128;          // hidden dim
constexpr float LN_EPS = 1e-5f;

typedef __attribute__((ext_vector_type(2))) float v2f;
typedef __attribute__((ext_vector_type(8))) float v8f;

// ---------------------------------------------------------------------------
// Kernel 1: T = X @ W1 + b1      X:[N,256]  W1:[256,128]  T:[N,128]
// Block = 256 threads (8 waves). Each block: 16-row panel; wave w -> cols
// [16w, 16w+16). fp32 WMMA 16x16x4, K-loop of 64 steps.
// ---------------------------------------------------------------------------
__global__ __launch_bounds__(256) void gemm1_wmma(
    const float* __restrict__ X, const float* __restrict__ W1,
    const float* __restrict__ b1, float* __restrict__ T, int N)
{
  constexpr int SA = D + 4;            // LDS row stride pad (bank spread)
  __shared__ float sA[16 * SA];        // 16x256 panel of X
  const int rowBase = blockIdx.x * 16;
  const bool full = (rowBase + 16 <= N);
  const int tid = threadIdx.x, wave = tid >> 5, lane = tid & 31;

  if (full) {
    const float* src = X + (size_t)rowBase * D;
    for (int i = tid; i < 16 * D; i += 256) {
      int r = i >> 8, c = i & (D - 1);
      sA[r * SA + c] = src[i];
    }
  } else {
    for (int i = tid; i < 16 * D; i += 256) {
      int r = i >> 8, c = i & (D - 1);
      int gr = min(rowBase + r, N - 1);
      sA[r * SA + c] = X[(size_t)gr * D + c];
    }
  }
  __syncthreads();

  const int col0 = wave * 16;
  const int l15  = lane & 15;          // A: row M   B: col N
  const int koff = (lane >> 4) << 1;   // lane half selects K {0,1} vs {2,3}

  v8f acc = {};
  #pragma unroll 4
  for (int k = 0; k < D; k += 4) {
    const int kk = k + koff;
    v2f a, b;
    a.x = sA[l15 * SA + kk];
    a.y = sA[l15 * SA + kk + 1];
    b.x = W1[(size_t)kk * H + col0 + l15];
    b.y = W1[(size_t)(kk + 1) * H + col0 + l15];
    acc = __builtin_amdgcn_wmma_f32_16x16x4_f32(false, a, false, b,
                                                (short)0, acc, false, false);
  }

  const int gcol = col0 + l15;
  const float bias = b1[gcol];
  const int mofs = (lane < 16) ? 0 : 8;
  if (full) {
    float* dst = T + (size_t)(rowBase + mofs) * H + gcol;
    #pragma unroll
    for (int r = 0; r < 8; ++r)
      dst[(size_t)r * H] = acc[r] + bias;
  } else {
    #pragma unroll
    for (int r = 0; r < 8; ++r) {
      int grow = rowBase + r + mofs;
      if (grow < N) T[(size_t)grow * H + gcol] = acc[r] + bias;
    }
  }
}

// ---------------------------------------------------------------------------
// Kernel 2: zero the aggregation buffer (must re-init every call).
// ---------------------------------------------------------------------------
__global__ void zero_f32(float* __restrict__ p, int n)
{
  int i = blockIdx.x * blockDim.x + threadIdx.x;
  if (i < n) p[i] = 0.f;
}

// ---------------------------------------------------------------------------
// Kernel 3: edge scatter. One wave32 per edge: coalesced 512B gather of
// transformed[src], scale by edge weight, 4x f32 hardware atomics per lane
// into aggregated[dst]. Both buffers are L2-resident (25.6 MB each).
// ---------------------------------------------------------------------------
__global__ __launch_bounds__(256) void edge_scatter(
    const float* __restrict__ T, const float* __restrict__ evals,
    const int* __restrict__ erows, const int* __restrict__ ecols,
    float* __restrict__ Agg, int E)
{
  const int wid  = (int)((blockIdx.x * (unsigned)blockDim.x + threadIdx.x) >> 5);
  const int lane = threadIdx.x & 31;
  if (wid >= E) return;
  const int src = ecols[wid];
  const int dst = erows[wid];
  const float w = evals[wid];
  const float4 v = ((const float4*)(T + (size_t)src * H))[lane];
  float* ap = Agg + (size_t)dst * H + lane * 4;
  unsafeAtomicAdd(ap + 0, w * v.x);     // -> global_atomic_add_f32 (no return)
  unsafeAtomicAdd(ap + 1, w * v.y);
  unsafeAtomicAdd(ap + 2, w * v.z);
  unsafeAtomicAdd(ap + 3, w * v.w);
}

// ---------------------------------------------------------------------------
// Kernel 4: Out = LayerNorm(Agg @ W2 + b2 + X) * gamma + beta
// Block = 512 threads (16 waves). Waves 0..15 each compute one 16x16 tile of
// the 16x256 panel via fp32 WMMA; bias+residual staged in LDS; then wave w
// LayerNorms panel row w with a wave32 shuffle reduction.
// ---------------------------------------------------------------------------
__global__ __launch_bounds__(512) void gemm2_ln_wmma(
    const float* __restrict__ Agg, const float* __restrict__ W2,
    const float* __restrict__ b2, const float* __restrict__ X,
    const float* __restrict__ gamma, const float* __restrict__ beta,
    float* __restrict__ Out, int N)
{
  constexpr int SA = H + 4;            // 132: bank-spread pad
  constexpr int SR = D + 1;            // 257: bank-spread pad
  __shared__ float sA[16 * SA];        // 16x128 panel of Agg  (~8.5 KB)
  __shared__ float sRes[16 * SR];      // 16x256 residual panel (~16.5 KB)
  const int rowBase = blockIdx.x * 16;
  const bool full = (rowBase + 16 <= N);
  const int tid = threadIdx.x, wave = tid >> 5, lane = tid & 31;

  if (full) {
    const float* src = Agg + (size_t)rowBase * H;
    for (int i = tid; i < 16 * H; i += 512) {
      int r = i >> 7, c = i & (H - 1);
      sA[r * SA + c] = src[i];
    }
  } else {
    for (int i = tid; i < 16 * H; i += 512) {
      int r = i >> 7, c = i & (H - 1);
      int gr = min(rowBase + r, N - 1);
      sA[r * SA + c] = Agg[(size_t)gr * H + c];
    }
  }
  __syncthreads();

  const int col0 = wave * 16;
  const int l15  = lane & 15;
  const int koff = (lane >> 4) << 1;

  v8f acc = {};
  #pragma unroll 4
  for (int k = 0; k < H; k += 4) {
    const int kk = k + koff;
    v2f a, b;
    a.x = sA[l15 * SA + kk];
    a.y = sA[l15 * SA + kk + 1];
    b.x = W2[(size_t)kk * D + col0 + l15];
    b.y = W2[(size_t)(kk + 1) * D + col0 + l15];
    acc = __builtin_amdgcn_wmma_f32_16x16x4_f32(false, a, false, b,
                                                (short)0, acc, false, false);
  }

  const int gcol = col0 + l15;
  const float bias = b2[gcol];
  const int mofs = (lane < 16) ? 0 : 8;
  if (full) {
    const float* xr = X + (size_t)(rowBase + mofs) * D + gcol;
    #pragma unroll
    for (int r = 0; r < 8; ++r)
      sRes[(r + mofs) * SR + gcol] = acc[r] + bias + xr[(size_t)r * D];
  } else {
    #pragma unroll
    for (int r = 0; r < 8; ++r) {
      int m = r + mofs;
      int gr = min(rowBase + m, N - 1);
      sRes[m * SR + gcol] = acc[r] + bias + X[(size_t)gr * D + gcol];
    }
  }
  __syncthreads();

  // Wave w LayerNorms panel row w (256 values, 8 per lane).
  const int m = wave;
  float vals[8];
  float s = 0.f, ss = 0.f;
  #pragma unroll
  for (int i = 0; i < 8; ++i) {
    float v = sRes[m * SR + lane + i * 32];
    vals[i] = v; s += v; ss += v * v;
  }
  #pragma unroll
  for (int off = 16; off > 0; off >>= 1) {
    s  += __shfl_xor(s,  off, 32);
    ss += __shfl_xor(ss, off, 32);
  }
  const float mu  = s  * (1.f / 256.f);
  const float var = ss * (1.f / 256.f) - mu * mu;   // biased variance
  const float inv = rsqrtf(var + LN_EPS);
  const int grow = rowBase + m;
  if (grow < N) {
    float* orow = Out + (size_t)grow * D;
    #pragma unroll
    for (int i = 0; i < 8; ++i) {
      int c = lane + i * 32;
      orow[c] = gamma[c] * (vals[i] - mu) * inv + beta[c];
    }
  }
}

// ---------------------------------------------------------------------------
extern "C" void kernel_launch(void* const* d_in, const int* in_sizes, int n_in,
                              void* d_out, int out_size, void* d_ws, size_t ws_size,
                              hipStream_t stream)
{
  const float* X     = (const float*)d_in[0];
  const float* W1    = (const float*)d_in[1];
  const float* b1    = (const float*)d_in[2];
  const float* W2    = (const float*)d_in[3];
  const float* b2    = (const float*)d_in[4];
  const float* gamma = (const float*)d_in[5];
  const float* beta  = (const float*)d_in[6];
  const float* evals = (const float*)d_in[7];
  const int*   erows = (const int*)d_in[8];
  const int*   ecols = (const int*)d_in[9];

  const int N = in_sizes[0] / D;       // 50000
  const int E = in_sizes[7];           // 800000

  float* T   = (float*)d_ws;                       // [N,H] transformed
  float* Agg = T + (size_t)N * H;                  // [N,H] aggregated
  float* Out = (float*)d_out;                      // [N,D]

  const int nNH = N * H;
  zero_f32<<<(nNH + 255) / 256, 256, 0, stream>>>(Agg, nNH);
  gemm1_wmma<<<(N + 15) / 16, 256, 0, stream>>>(X, W1, b1, T, N);
  edge_scatter<<<(E + 7) / 8, 256, 0, stream>>>(T, evals, erows, ecols, Agg, E);
  gemm2_ln_wmma<<<(N + 15) / 16, 512, 0, stream>>>(Agg, W2, b2, X, gamma, beta, Out, N);
}